// splineN_59098749993116
// MI455X (gfx1250) — compile-verified
//
#include <hip/hip_runtime.h>
#include <hip/hip_bf16.h>
#include <math.h>

// ---------------------------------------------------------------------------
// SplineConv GNN, 2 layers: 61 -> 16 (ELU) -> 2 (log_softmax), mean aggr.
// f32 throughout (matches reference precision). Node transforms of layer 1
// use V_WMMA_F32_16X16X4_F32; edge scatter uses L2-resident f32 atomics.
// ---------------------------------------------------------------------------

typedef float v2f __attribute__((ext_vector_type(2)));
typedef float v8f __attribute__((ext_vector_type(8)));

#define F_IN  61
#define F_HID 16
#define F_OUT 2

// ---------------------------------------------------------------------------
// Zero a contiguous float region (accumulators must be re-zeroed every call).
__global__ void zero_kernel(float* __restrict__ p, int n) {
    int i = blockIdx.x * blockDim.x + threadIdx.x;
    if (i < n) p[i] = 0.0f;
}

// ---------------------------------------------------------------------------
// Layer-1 node transforms via FP32 WMMA.
//   h0 = x @ W1[0], h1 = x @ W1[1], xr = x @ root1     (all [N,16])
// One wave per 16-node tile. K = 61 padded to 64, 16 steps of 16x16x4.
// A frag (16x4 f32): lane m=l&15 is row; lo half-wave holds {K,K+1},
//   hi half-wave holds {K+2,K+3}  (per ISA 7.12.2, 32-bit A 16x4).
// B frag (4x16 f32): lane n=l&15 is column; same K split as A.
// C/D (16x16 f32): element j -> row j + 8*(l>>4), col l&15.
//
// No per-lane predicated loads: row index is clamped (loads always valid,
// stores guarded), and the single partial K-step (k=60..63) uses
// unconditional loads from clamped addresses + v_cndmask zero-selects.
__global__ __launch_bounds__(128)
void node_transform1_wmma(const float* __restrict__ x,     // [N,61]
                          const float* __restrict__ W1,    // [2,61,16]
                          const float* __restrict__ root1, // [61,16]
                          float* __restrict__ h0,          // [N,16]
                          float* __restrict__ h1,          // [N,16]
                          float* __restrict__ xr,          // [N,16]
                          int n_nodes, int n_tiles) {
    const int lane = threadIdx.x & 31;
    const int wave = (blockIdx.x * (blockDim.x >> 5)) + (threadIdx.x >> 5);
    if (wave >= n_tiles) return;           // wave-uniform: EXEC stays all-1s

    const int m  = lane & 15;              // A row / B,D column
    const int hi = lane >> 4;              // half-wave selects K pair
    int node = wave * 16 + m;
    if (node > n_nodes - 1) node = n_nodes - 1;   // clamp: loads always valid

    const float* __restrict__ xrow = x + (long long)node * F_IN;
    const float* __restrict__ Wa   = W1;                 // W1[0]: [61,16]
    const float* __restrict__ Wb   = W1 + F_IN * F_HID;  // W1[1]

    // ---- preload all 16 A fragments (steps 0..14 fully in-bounds) ----
    v2f afrag[16];
    #pragma unroll
    for (int kb = 0; kb < 15; ++kb) {
        const int k = kb * 4 + hi * 2;     // max 59 < 61
        afrag[kb].x = xrow[k];
        afrag[kb].y = xrow[k + 1];
    }
    {   // step 15: lo half k=60 valid, k=61 pad; hi half k=62,63 pad
        const float a60 = xrow[60];        // unconditional, always valid
        afrag[15].x = hi ? 0.0f : a60;     // v_cndmask, no branch
        afrag[15].y = 0.0f;
    }

    v8f c0 = {}; v8f c1 = {}; v8f cr = {};

    // ---- main K loop: 6 B loads + 3 WMMAs per step ----
    #pragma unroll
    for (int kb = 0; kb < 15; ++kb) {
        const int k = kb * 4 + hi * 2;
        v2f b0, b1v, br;
        b0.x  = Wa[(k    ) * F_HID + m];
        b0.y  = Wa[(k + 1) * F_HID + m];
        b1v.x = Wb[(k    ) * F_HID + m];
        b1v.y = Wb[(k + 1) * F_HID + m];
        br.x  = root1[(k    ) * F_HID + m];
        br.y  = root1[(k + 1) * F_HID + m];

        c0 = __builtin_amdgcn_wmma_f32_16x16x4_f32(false, afrag[kb], false, b0,
                                                   (short)0, c0, false, false);
        c1 = __builtin_amdgcn_wmma_f32_16x16x4_f32(false, afrag[kb], false, b1v,
                                                   (short)0, c1, false, false);
        cr = __builtin_amdgcn_wmma_f32_16x16x4_f32(false, afrag[kb], false, br,
                                                   (short)0, cr, false, false);
    }
    {   // K tail (step 15): only k=60 row of B contributes, lo half-wave
        v2f b0, b1v, br;
        const float w0 = Wa[60 * F_HID + m];
        const float w1 = Wb[60 * F_HID + m];
        const float wr = root1[60 * F_HID + m];
        b0.x  = hi ? 0.0f : w0;   b0.y  = 0.0f;
        b1v.x = hi ? 0.0f : w1;   b1v.y = 0.0f;
        br.x  = hi ? 0.0f : wr;   br.y  = 0.0f;

        c0 = __builtin_amdgcn_wmma_f32_16x16x4_f32(false, afrag[15], false, b0,
                                                   (short)0, c0, false, false);
        c1 = __builtin_amdgcn_wmma_f32_16x16x4_f32(false, afrag[15], false, b1v,
                                                   (short)0, c1, false, false);
        cr = __builtin_amdgcn_wmma_f32_16x16x4_f32(false, afrag[15], false, br,
                                                   (short)0, cr, false, false);
    }

    #pragma unroll
    for (int j = 0; j < 8; ++j) {
        const int row = wave * 16 + j + 8 * hi;
        if (row < n_nodes) {
            h0[(long long)row * F_HID + m] = c0[j];
            h1[(long long)row * F_HID + m] = c1[j];
            xr[(long long)row * F_HID + m] = cr[j];
        }
    }
}

// ---------------------------------------------------------------------------
// Edge pass 1: thread per (edge, feature). 16 consecutive lanes share an edge
// -> 64B-contiguous gathers of h0/h1[src] and 64B-contiguous atomic scatters.
// Also builds cnt (in-degree) once; it is reused by layer 2.
__global__ __launch_bounds__(256)
void edge_pass1(const long long* __restrict__ src,
                const long long* __restrict__ dst,
                const float* __restrict__ u,        // [E]
                const float* __restrict__ h0,
                const float* __restrict__ h1,
                float* __restrict__ summed1,        // [N,16]
                float* __restrict__ cnt,            // [N]
                long long n_edges) {
    const long long tid = (long long)blockIdx.x * blockDim.x + threadIdx.x;
    const long long e = tid >> 4;
    if (e >= n_edges) return;
    const int f = (int)(tid & 15);

    const long long s = src[e];
    const long long d = dst[e];
    const float w1 = u[e];
    const float w0 = 1.0f - w1;

    const float msg = w0 * h0[s * F_HID + f] + w1 * h1[s * F_HID + f];
    atomicAdd(&summed1[d * F_HID + f], msg);
    if (f == 0) atomicAdd(&cnt[d], 1.0f);
}

// ---------------------------------------------------------------------------
// Epilogue 1 fused with layer-2 node transforms:
//   act = elu(summed1/max(cnt,1) + xr + b1)
//   g0 = act @ W2[0], g1 = act @ W2[1], rr = act @ root2    ([N,2] each)
// act never touches memory.
__global__ __launch_bounds__(256)
void finish1_transform2(const float* __restrict__ summed1,
                        const float* __restrict__ cnt,
                        const float* __restrict__ xr,
                        const float* __restrict__ b1,    // [16]
                        const float* __restrict__ W2,    // [2,16,2]
                        const float* __restrict__ root2, // [16,2]
                        float* __restrict__ g0,          // [N,2]
                        float* __restrict__ g1,          // [N,2]
                        float* __restrict__ rr,          // [N,2]
                        int n_nodes) {
    const int node = blockIdx.x * blockDim.x + threadIdx.x;
    if (node >= n_nodes) return;

    const float inv = 1.0f / fmaxf(cnt[node], 1.0f);
    float a00 = 0.f, a01 = 0.f, a10 = 0.f, a11 = 0.f, ar0 = 0.f, ar1 = 0.f;

    #pragma unroll
    for (int f = 0; f < F_HID; ++f) {
        float v = summed1[(long long)node * F_HID + f] * inv
                + xr[(long long)node * F_HID + f] + b1[f];
        v = (v > 0.0f) ? v : (expf(v) - 1.0f);          // ELU, alpha=1
        a00 += v * W2[f * 2 + 0];
        a01 += v * W2[f * 2 + 1];
        a10 += v * W2[F_HID * 2 + f * 2 + 0];
        a11 += v * W2[F_HID * 2 + f * 2 + 1];
        ar0 += v * root2[f * 2 + 0];
        ar1 += v * root2[f * 2 + 1];
    }
    g0[node * 2 + 0] = a00;  g0[node * 2 + 1] = a01;
    g1[node * 2 + 0] = a10;  g1[node * 2 + 1] = a11;
    rr[node * 2 + 0] = ar0;  rr[node * 2 + 1] = ar1;
}

// ---------------------------------------------------------------------------
// Edge pass 2: thread per edge, 2 features.
__global__ __launch_bounds__(256)
void edge_pass2(const long long* __restrict__ src,
                const long long* __restrict__ dst,
                const float* __restrict__ u,
                const float* __restrict__ g0,
                const float* __restrict__ g1,
                float* __restrict__ summed2,    // [N,2]
                long long n_edges) {
    const long long e = (long long)blockIdx.x * blockDim.x + threadIdx.x;
    if (e >= n_edges) return;
    const long long s = src[e];
    const long long d = dst[e];
    const float w1 = u[e];
    const float w0 = 1.0f - w1;
    atomicAdd(&summed2[d * 2 + 0], w0 * g0[s * 2 + 0] + w1 * g1[s * 2 + 0]);
    atomicAdd(&summed2[d * 2 + 1], w0 * g0[s * 2 + 1] + w1 * g1[s * 2 + 1]);
}

// ---------------------------------------------------------------------------
// Final: mean + root + bias, then 2-class log_softmax.
__global__ __launch_bounds__(256)
void final_logsoftmax(const float* __restrict__ summed2,
                      const float* __restrict__ cnt,
                      const float* __restrict__ rr,
                      const float* __restrict__ b2,   // [2]
                      float* __restrict__ out,        // [N,2]
                      int n_nodes) {
    const int node = blockIdx.x * blockDim.x + threadIdx.x;
    if (node >= n_nodes) return;
    const float inv = 1.0f / fmaxf(cnt[node], 1.0f);
    const float v0 = summed2[node * 2 + 0] * inv + rr[node * 2 + 0] + b2[0];
    const float v1 = summed2[node * 2 + 1] * inv + rr[node * 2 + 1] + b2[1];
    const float mx  = fmaxf(v0, v1);
    const float lse = mx + logf(expf(v0 - mx) + expf(v1 - mx));
    out[node * 2 + 0] = v0 - lse;
    out[node * 2 + 1] = v1 - lse;
}

// ---------------------------------------------------------------------------
extern "C" void kernel_launch(void* const* d_in, const int* in_sizes, int n_in,
                              void* d_out, int out_size, void* d_ws, size_t ws_size,
                              hipStream_t stream) {
    // Inputs (setup_inputs order):
    // 0:x[N,61] f32  1:edge_index[2,E] i64  2:edge_attr[E,1] f32  3:dropout i32
    // 4:W1[2,61,16]  5:root1[61,16]  6:b1[16]  7:W2[2,16,2]  8:root2[16,2]  9:b2[2]
    const float*     x      = (const float*)d_in[0];
    const long long* eidx   = (const long long*)d_in[1];
    const float*     eattr  = (const float*)d_in[2];
    const float*     W1     = (const float*)d_in[4];
    const float*     root1  = (const float*)d_in[5];
    const float*     b1     = (const float*)d_in[6];
    const float*     W2     = (const float*)d_in[7];
    const float*     root2  = (const float*)d_in[8];
    const float*     b2     = (const float*)d_in[9];

    const int       n_nodes = in_sizes[0] / F_IN;
    const long long n_edges = (long long)in_sizes[2];      // edge_attr count
    const long long* src = eidx;
    const long long* dst = eidx + n_edges;

    // Workspace layout (floats):
    //   h0[N*16] h1[N*16] xr[N*16] g0[N*2] g1[N*2] rr[N*2]
    //   | summed1[N*16] cnt[N] summed2[N*2]   <- contiguous zero region
    float* ws = (float*)d_ws;
    const long long N = n_nodes;
    float* h0      = ws;
    float* h1      = h0 + N * F_HID;
    float* xr      = h1 + N * F_HID;
    float* g0      = xr + N * F_HID;
    float* g1      = g0 + N * 2;
    float* rr      = g1 + N * 2;
    float* summed1 = rr + N * 2;
    float* cnt     = summed1 + N * F_HID;
    float* summed2 = cnt + N;
    const int zero_count = (int)(N * F_HID + N + N * 2);   // summed1|cnt|summed2

    // 1) zero accumulators
    zero_kernel<<<(zero_count + 255) / 256, 256, 0, stream>>>(summed1, zero_count);

    // 2) layer-1 node transforms via FP32 WMMA (one wave per 16-node tile)
    const int n_tiles = (n_nodes + 15) / 16;
    const int waves_per_block = 4;                          // blockDim 128
    const int blocks_t1 = (n_tiles + waves_per_block - 1) / waves_per_block;
    node_transform1_wmma<<<blocks_t1, 32 * waves_per_block, 0, stream>>>(
        x, W1, root1, h0, h1, xr, n_nodes, n_tiles);

    // 3) edge pass 1 (thread per edge-feature)
    const long long t1 = n_edges * F_HID;
    edge_pass1<<<(int)((t1 + 255) / 256), 256, 0, stream>>>(
        src, dst, eattr, h0, h1, summed1, cnt, n_edges);

    // 4) epilogue-1 fused with layer-2 transforms
    finish1_transform2<<<(n_nodes + 255) / 256, 256, 0, stream>>>(
        summed1, cnt, xr, b1, W2, root2, g0, g1, rr, n_nodes);

    // 5) edge pass 2
    edge_pass2<<<(int)((n_edges + 255) / 256), 256, 0, stream>>>(
        src, dst, eattr, g0, g1, summed2, n_edges);

    // 6) final mean + root + bias + log_softmax
    final_logsoftmax<<<(n_nodes + 255) / 256, 256, 0, stream>>>(
        summed2, cnt, rr, b2, (float*)d_out, n_nodes);
}